// SPRGCN_88648124990792
// MI455X (gfx1250) — compile-verified
//
#include <hip/hip_runtime.h>
#include <hip/hip_bf16.h>

typedef __attribute__((ext_vector_type(2))) float v2f;
typedef __attribute__((ext_vector_type(8))) float v8f;

#define N_NODES   100000
#define N_EDGES   1600000
#define IN_DIM    128
#define HID       128
#define N_CLS     10
#define N_GRAPHS  128

// ---------------------------------------------------------------------------
// deg[i] = 1.0 (self loop contribution)
__global__ void k_init_deg(float* __restrict__ deg, int n) {
    int i = blockIdx.x * blockDim.x + threadIdx.x;
    if (i < n) deg[i] = 1.0f;
}

// deg[col[e]] += 1 for every real edge
__global__ void k_deg_accum(const long long* __restrict__ col,
                            float* __restrict__ deg, int nE) {
    int e = blockIdx.x * blockDim.x + threadIdx.x;
    if (e < nE) atomicAdd(&deg[(int)col[e]], 1.0f);
}

// deg -> rsqrt(deg)  (deg >= 1 always, so no zero guard needed)
__global__ void k_dinv(float* __restrict__ deg, int n) {
    int i = blockIdx.x * blockDim.x + threadIdx.x;
    if (i < n) deg[i] = rsqrtf(deg[i]);
}

// ---------------------------------------------------------------------------
// H[M x 128] = A[M x 128] @ W[128 x 128] using V_WMMA_F32_16X16X4_F32.
// One wave computes one 16x16 output tile; 8 column tiles (N=128).
__global__ void k_gemm_wmma_f32(const float* __restrict__ A,
                                const float* __restrict__ W,
                                float* __restrict__ H, int M) {
    const int ntN = HID / 16;                       // 8 column tiles
    int wave = (blockIdx.x * blockDim.x + threadIdx.x) >> 5;
    int lane = threadIdx.x & 31;
    int tm = wave / ntN;
    int tn = wave - tm * ntN;
    if (tm * 16 >= M) return;                       // wave-uniform: EXEC all-1s inside

    const int m0 = tm * 16;
    const int n0 = tn * 16;
    const int l15 = lane & 15;
    const int khalf = (lane >> 4) << 1;             // 0 for lanes 0-15, 2 for 16-31

    const float* arow = A + (size_t)(m0 + l15) * HID;   // A: M = lane&15
    const float* wcol = W + (n0 + l15);                 // B: N = lane&15

    v8f acc = {};
    #pragma unroll
    for (int k = 0; k < IN_DIM; k += 4) {
        v2f a, b;
        a.x = arow[k + khalf];                      // A vgpr0: K = khalf
        a.y = arow[k + khalf + 1];                  // A vgpr1: K = khalf+1
        b.x = wcol[(size_t)(k + khalf) * HID];      // B vgpr0: K = khalf
        b.y = wcol[(size_t)(k + khalf + 1) * HID];  // B vgpr1: K = khalf+1
        acc = __builtin_amdgcn_wmma_f32_16x16x4_f32(
            /*neg_a=*/false, a, /*neg_b=*/false, b,
            /*c_mod=*/(short)0, acc, /*reuse_a=*/false, /*reuse_b=*/false);
    }

    // D layout: vgpr r -> row m0 + r + (lane>=16 ? 8 : 0), col n0 + (lane&15)
    int rbase = m0 + ((lane >> 4) << 3);
    float* out = H + (size_t)rbase * HID + (n0 + l15);
    #pragma unroll
    for (int r = 0; r < 8; ++r) out[(size_t)r * HID] = acc[r];
}

// ---------------------------------------------------------------------------
// agg[i, :] = h[i, :] * dinv[i]^2   (self-loop term; also initializes agg)
__global__ void k_selfloop_init(const float* __restrict__ h,
                                const float* __restrict__ dinv,
                                float* __restrict__ agg, int n) {
    int idx4 = blockIdx.x * blockDim.x + threadIdx.x;   // over n*32 float4 chunks
    if (idx4 >= n * (HID / 4)) return;
    int node = idx4 >> 5;
    float d = dinv[node];
    float s = d * d;
    float4 v = ((const float4*)h)[idx4];
    float4 o; o.x = v.x * s; o.y = v.y * s; o.z = v.z * s; o.w = v.w * s;
    ((float4*)agg)[idx4] = o;
}

// One wave per edge: agg[col, :] += h[row, :] * dinv[row]*dinv[col]
__global__ void k_edge_agg(const long long* __restrict__ rows,
                           const long long* __restrict__ cols,
                           const float* __restrict__ dinv,
                           const float* __restrict__ h,
                           float* __restrict__ agg, int nE) {
    int wid = (blockIdx.x * blockDim.x + threadIdx.x) >> 5;
    int lane = threadIdx.x & 31;
    if (wid >= nE) return;
    int r = (int)rows[wid];
    int c = (int)cols[wid];
    float nrm = dinv[r] * dinv[c];
    float4 v = ((const float4*)(h + (size_t)r * HID))[lane];
    float* o = agg + (size_t)c * HID + lane * 4;
    atomicAdd(o + 0, v.x * nrm);
    atomicAdd(o + 1, v.y * nrm);
    atomicAdd(o + 2, v.z * nrm);
    atomicAdd(o + 3, v.w * nrm);
}

// agg = relu(agg + b)
__global__ void k_bias_relu(float* __restrict__ agg,
                            const float* __restrict__ b, int n) {
    int idx4 = blockIdx.x * blockDim.x + threadIdx.x;   // over n*32 float4 chunks
    if (idx4 >= n * (HID / 4)) return;
    float4 v = ((const float4*)agg)[idx4];
    float4 bb = ((const float4*)b)[idx4 & 31];
    v.x = fmaxf(v.x + bb.x, 0.0f);
    v.y = fmaxf(v.y + bb.y, 0.0f);
    v.z = fmaxf(v.z + bb.z, 0.0f);
    v.w = fmaxf(v.w + bb.w, 0.0f);
    ((float4*)agg)[idx4] = v;
}

// ---------------------------------------------------------------------------
__global__ void k_zero_pool(float* __restrict__ pool, float* __restrict__ cnt) {
    int i = blockIdx.x * blockDim.x + threadIdx.x;
    if (i < N_GRAPHS * HID) pool[i] = 0.0f;
    if (i < N_GRAPHS) cnt[i] = 0.0f;
}

// One wave per node: pool[batch[i], :] += h[i, :];  cnt[batch[i]] += 1
__global__ void k_pool_agg(const long long* __restrict__ batch,
                           const float* __restrict__ h,
                           float* __restrict__ pool,
                           float* __restrict__ cnt, int n) {
    int wid = (blockIdx.x * blockDim.x + threadIdx.x) >> 5;
    int lane = threadIdx.x & 31;
    if (wid >= n) return;
    int g = (int)batch[wid];
    float4 v = ((const float4*)(h + (size_t)wid * HID))[lane];
    float* p = pool + (size_t)g * HID + lane * 4;
    atomicAdd(p + 0, v.x);
    atomicAdd(p + 1, v.y);
    atomicAdd(p + 2, v.z);
    atomicAdd(p + 3, v.w);
    if (lane == 0) atomicAdd(&cnt[g], 1.0f);
}

// out[g, c] = (sum_k pool[g,k] * Wlin[k,c]) / max(cnt[g],1) + blin[c]
__global__ void k_final_linear(const float* __restrict__ pool,
                               const float* __restrict__ cnt,
                               const float* __restrict__ Wlin,
                               const float* __restrict__ blin,
                               float* __restrict__ out) {
    int idx = blockIdx.x * blockDim.x + threadIdx.x;
    if (idx >= N_GRAPHS * N_CLS) return;
    int g = idx / N_CLS;
    int c = idx - g * N_CLS;
    float inv = 1.0f / fmaxf(cnt[g], 1.0f);
    const float* pg = pool + (size_t)g * HID;
    float s = 0.0f;
    #pragma unroll 8
    for (int k = 0; k < HID; ++k) s += pg[k] * Wlin[(size_t)k * N_CLS + c];
    out[idx] = s * inv + blin[c];
}

// ---------------------------------------------------------------------------
static inline size_t align_up(size_t x, size_t a) { return (x + a - 1) & ~(a - 1); }

extern "C" void kernel_launch(void* const* d_in, const int* in_sizes, int n_in,
                              void* d_out, int out_size, void* d_ws, size_t ws_size,
                              hipStream_t stream) {
    (void)in_sizes; (void)n_in; (void)out_size; (void)ws_size;

    const float*     x    = (const float*)d_in[0];
    const long long* ei   = (const long long*)d_in[1];   // [2, N_EDGES] int64
    const long long* batch= (const long long*)d_in[2];   // [N_NODES] int64
    const float*     W1   = (const float*)d_in[3];
    const float*     b1   = (const float*)d_in[4];
    const float*     W2   = (const float*)d_in[5];
    const float*     b2   = (const float*)d_in[6];
    const float*     Wlin = (const float*)d_in[7];
    const float*     blin = (const float*)d_in[8];
    float* out = (float*)d_out;

    const long long* rows = ei;             // edge_index[0] = source
    const long long* cols = ei + N_EDGES;   // edge_index[1] = destination

    // Workspace layout
    char* ws = (char*)d_ws;
    size_t off = 0;
    float* dinv = (float*)(ws + off); off = align_up(off + (size_t)N_NODES * 4, 1024);
    float* hbuf = (float*)(ws + off); off = align_up(off + (size_t)N_NODES * HID * 4, 1024);
    float* aggb = (float*)(ws + off); off = align_up(off + (size_t)N_NODES * HID * 4, 1024);
    float* pool = (float*)(ws + off); off = align_up(off + (size_t)N_GRAPHS * HID * 4, 1024);
    float* cnt  = (float*)(ws + off); off = align_up(off + (size_t)N_GRAPHS * 4, 1024);

    const int T = 256;
    const int nodeBlk   = (N_NODES + T - 1) / T;
    const int edgeBlk   = (N_EDGES + T - 1) / T;
    const int featBlk   = ((N_NODES * (HID / 4)) + T - 1) / T;      // float4 chunks
    const int edgeWaveBlk = ((N_EDGES * 32) + T - 1) / T;           // wave per edge
    const int nodeWaveBlk = ((N_NODES * 32) + T - 1) / T;           // wave per node
    const int gemmWaves = (N_NODES / 16) * (HID / 16);              // 50000 waves
    const int gemmBlk   = ((gemmWaves * 32) + T - 1) / T;

    // --- normalization coefficients ---
    k_init_deg <<<nodeBlk, T, 0, stream>>>(dinv, N_NODES);
    k_deg_accum<<<edgeBlk, T, 0, stream>>>(cols, dinv, N_EDGES);
    k_dinv     <<<nodeBlk, T, 0, stream>>>(dinv, N_NODES);

    // --- layer 1: h = x@W1 ; agg = D^-1/2 A D^-1/2 h ; relu(agg + b1) ---
    k_gemm_wmma_f32<<<gemmBlk, T, 0, stream>>>(x, W1, hbuf, N_NODES);
    k_selfloop_init<<<featBlk, T, 0, stream>>>(hbuf, dinv, aggb, N_NODES);
    k_edge_agg     <<<edgeWaveBlk, T, 0, stream>>>(rows, cols, dinv, hbuf, aggb, N_EDGES);
    k_bias_relu    <<<featBlk, T, 0, stream>>>(aggb, b1, N_NODES);

    // --- layer 2 ---
    k_gemm_wmma_f32<<<gemmBlk, T, 0, stream>>>(aggb, W2, hbuf, N_NODES);
    k_selfloop_init<<<featBlk, T, 0, stream>>>(hbuf, dinv, aggb, N_NODES);
    k_edge_agg     <<<edgeWaveBlk, T, 0, stream>>>(rows, cols, dinv, hbuf, aggb, N_EDGES);
    k_bias_relu    <<<featBlk, T, 0, stream>>>(aggb, b2, N_NODES);

    // --- global mean pool + classifier head ---
    k_zero_pool   <<<(N_GRAPHS * HID + T - 1) / T, T, 0, stream>>>(pool, cnt);
    k_pool_agg    <<<nodeWaveBlk, T, 0, stream>>>(batch, aggb, pool, cnt, N_NODES);
    k_final_linear<<<(N_GRAPHS * N_CLS + T - 1) / T, T, 0, stream>>>(pool, cnt, Wlin, blin, out);
}